// Chocolate_68513318306430
// MI455X (gfx1250) — compile-verified
//
#include <hip/hip_runtime.h>
#include <hip/hip_bf16.h>

#define N_NODES 20000
#define N_EDGES 256000
#define N_MOLS  1000
#define HID     128
#define H2      256
#define EPS_F   1e-8f

typedef __attribute__((ext_vector_type(2))) float v2f;
typedef __attribute__((ext_vector_type(8))) float v8f;

__device__ __forceinline__ float silu_f(float x) {
    return x / (1.0f + __expf(-x));
}

// One K-sweep producing two 16x16 f32 tiles (rows 0-15 and 16-31 of a 32-row
// A tile in LDS) against one B panel in global memory.  B fragment is loaded
// once per K-step and reused for both M-tiles.
// Assumed V_WMMA_F32_16X16X4_F32 layouts (ISA 7.12.2):
//   A: lane(0-15)=M, vgpr0/1 = K{0,1}; lane(16-31)=M, vgpr0/1 = K{2,3}
//   B: mirrored (lane half selects K pair, laneM selects N)
//   C/D: vgpr i -> M = i + 8*half, N = n0 + laneM
__device__ __forceinline__ void wmma_pair_k(
    const float* __restrict__ Bw, int ldb,
    const float* sA, int lda, int K,
    int n0, int laneM, int half,
    v8f& acc0, v8f& acc1)
{
    const float* a0p = sA + laneM * lda;
    const float* a1p = sA + (16 + laneM) * lda;
#pragma unroll 4
    for (int k0 = 0; k0 < K; k0 += 4) {
        const int kk = k0 + 2 * half;
        v2f b; b.x = Bw[kk * ldb + n0 + laneM];
               b.y = Bw[(kk + 1) * ldb + n0 + laneM];
        v2f a0; a0.x = a0p[kk]; a0.y = a0p[kk + 1];
        v2f a1; a1.x = a1p[kk]; a1.y = a1p[kk + 1];
        acc0 = __builtin_amdgcn_wmma_f32_16x16x4_f32(false, a0, false, b,
                                                     (short)0, acc0, false, false);
        acc1 = __builtin_amdgcn_wmma_f32_16x16x4_f32(false, a1, false, b,
                                                     (short)0, acc1, false, false);
    }
}

// ---------------------------------------------------------------- init ----
__global__ void init_x_kernel(const int* __restrict__ z,
                              const float* __restrict__ emb,
                              float* __restrict__ x)
{
    int idx = blockIdx.x * blockDim.x + threadIdx.x;
    if (idx < N_NODES * HID)
        x[idx] = emb[z[idx >> 7] * HID + (idx & 127)];
}

__global__ void init_v_kernel(const float* __restrict__ pos,
                              float* __restrict__ v)
{
    int idx = blockIdx.x * blockDim.x + threadIdx.x;
    if (idx < N_NODES * 3 * HID) {
        int node = idx / (3 * HID);
        int d    = (idx >> 7) % 3;
        v[idx]   = pos[node * 3 + d];
    }
}

__global__ void geom_kernel(const float* __restrict__ pos,
                            const int* __restrict__ ei,
                            float* __restrict__ dist,
                            float* __restrict__ dir)
{
    int e = blockIdx.x * blockDim.x + threadIdx.x;
    if (e < N_EDGES) {
        int r = ei[e];
        int c = ei[N_EDGES + e];
        float dx = pos[c * 3 + 0] - pos[r * 3 + 0];
        float dy = pos[c * 3 + 1] - pos[r * 3 + 1];
        float dz = pos[c * 3 + 2] - pos[r * 3 + 2];
        float d  = sqrtf(dx * dx + dy * dy + dz * dz);
        float inv = 1.0f / fmaxf(d, EPS_F);
        dist[e]        = d;
        dir[e * 3 + 0] = dx * inv;
        dir[e * 3 + 1] = dy * inv;
        dir[e * 3 + 2] = dz * inv;
    }
}

__global__ void zero_kernel(float* __restrict__ p, int n)
{
    int idx = blockIdx.x * blockDim.x + threadIdx.x;
    if (idx < n) p[idx] = 0.0f;
}

// ------------------------------------------------------- edge message ----
// 32 edges per block, 256 threads (8 waves).  Two fused 256-wide MLP layers
// entirely in LDS, then atomic scatter of scalar + vector messages.
#define SA_LD 260   // 260 % 64 == 4 -> 16-lane ds_load conflict free

__global__ void __launch_bounds__(256)
edge_kernel(const float* __restrict__ x,
            const int* __restrict__ ei,
            const float* __restrict__ dist,
            const float* __restrict__ dir,
            const float* __restrict__ W1, const float* __restrict__ b1,
            const float* __restrict__ W2, const float* __restrict__ b2,
            float* __restrict__ agg_s, float* __restrict__ agg_v)
{
    __shared__ float sA [32 * SA_LD];   // m_in  [32][257]
    __shared__ float sH1[32 * SA_LD];   // h1    [32][256]
    __shared__ float sH2[32 * SA_LD];   // h2    [32][256]
    __shared__ int   sRow[32], sCol[32];
    __shared__ float sDir[32][3];

    const int tid = threadIdx.x;
    const int e0  = blockIdx.x * 32;

    if (tid < 32) {
        int e = e0 + tid;
        sRow[tid] = ei[e];
        sCol[tid] = ei[N_EDGES + e];
        sA[tid * SA_LD + 256] = dist[e];
        sDir[tid][0] = dir[e * 3 + 0];
        sDir[tid][1] = dir[e * 3 + 1];
        sDir[tid][2] = dir[e * 3 + 2];
    }
    __syncthreads();

    // gather m_in = [x[row] | x[col] | dist]
    for (int idx = tid; idx < 32 * H2; idx += 256) {
        int e = idx >> 8, k = idx & 255;
        int node = (k < HID) ? sRow[e] : sCol[e];
        sA[e * SA_LD + k] = x[node * HID + (k & 127)];
    }
    __syncthreads();

    const int lane  = tid & 31;
    const int wave  = tid >> 5;
    const int laneM = lane & 15;
    const int half  = lane >> 4;

    // ---- stage 1: h1 = silu(m_in @ W1 + b1), K = 256 (+ dist rank-1) ----
    for (int nt = 0; nt < 2; ++nt) {
        const int n0 = (wave * 2 + nt) * 16;
        v8f acc0 = {}, acc1 = {};
        wmma_pair_k(W1, H2, sA, SA_LD, H2, n0, laneM, half, acc0, acc1);
        const float bD   = W1[256 * H2 + n0 + laneM];   // dist row of W1
        const float bias = b1[n0 + laneM];
#pragma unroll
        for (int i = 0; i < 8; ++i) {
            int m = i + 8 * half;
            float v0 = acc0[i] + sA[m * SA_LD + 256] * bD + bias;
            float v1 = acc1[i] + sA[(m + 16) * SA_LD + 256] * bD + bias;
            sH1[m * SA_LD + n0 + laneM]        = silu_f(v0);
            sH1[(m + 16) * SA_LD + n0 + laneM] = silu_f(v1);
        }
    }
    __syncthreads();

    // ---- stage 2: h2 = silu(h1 @ W2 + b2), K = 256 ----
    for (int nt = 0; nt < 2; ++nt) {
        const int n0 = (wave * 2 + nt) * 16;
        v8f acc0 = {}, acc1 = {};
        wmma_pair_k(W2, H2, sH1, SA_LD, H2, n0, laneM, half, acc0, acc1);
        const float bias = b2[n0 + laneM];
#pragma unroll
        for (int i = 0; i < 8; ++i) {
            int m = i + 8 * half;
            sH2[m * SA_LD + n0 + laneM]        = silu_f(acc0[i] + bias);
            sH2[(m + 16) * SA_LD + n0 + laneM] = silu_f(acc1[i] + bias);
        }
    }
    __syncthreads();

    // ---- scatter: agg_s += msg_scalar ; agg_v += gate * dir ----
    for (int idx = tid; idx < 32 * HID; idx += 256) {
        int e = idx >> 7, c = idx & 127;
        int r = sRow[e];
        float ms = sH2[e * SA_LD + HID + c];   // msg_scalar = h[:,128:]
        float g  = sH2[e * SA_LD + c];         // gate_vec   = h[:,:128]
        atomicAdd(&agg_s[r * HID + c], ms);
        atomicAdd(&agg_v[(r * 3 + 0) * HID + c], g * sDir[e][0]);
        atomicAdd(&agg_v[(r * 3 + 1) * HID + c], g * sDir[e][1]);
        atomicAdd(&agg_v[(r * 3 + 2) * HID + c], g * sDir[e][2]);
    }
}

// ------------------------------------------------------- node update ----
// 32 nodes per block: u = silu([x|agg_s] @ updW1 + b1); x += u @ updW2 + b2
#define SU_LD 132   // 132 % 64 == 4 -> conflict free

__global__ void __launch_bounds__(256)
node_kernel(float* __restrict__ x, const float* __restrict__ agg_s,
            const float* __restrict__ W1, const float* __restrict__ b1,
            const float* __restrict__ W2, const float* __restrict__ b2)
{
    __shared__ float sA[32 * SA_LD];    // [x | agg_s]  [32][256]
    __shared__ float sU[32 * SU_LD];    // u            [32][128]

    const int tid = threadIdx.x;
    const int nb0 = blockIdx.x * 32;

    for (int idx = tid; idx < 32 * H2; idx += 256) {
        int m = idx >> 8, k = idx & 255;
        int node = nb0 + m;
        sA[m * SA_LD + k] = (k < HID) ? x[node * HID + k]
                                      : agg_s[node * HID + (k - HID)];
    }
    __syncthreads();

    const int lane  = tid & 31;
    const int wave  = tid >> 5;
    const int laneM = lane & 15;
    const int half  = lane >> 4;
    const int n0    = wave * 16;      // 8 waves cover 128 output cols

    {   // stage 1: K = 256
        v8f acc0 = {}, acc1 = {};
        wmma_pair_k(W1, HID, sA, SA_LD, H2, n0, laneM, half, acc0, acc1);
        const float bias = b1[n0 + laneM];
#pragma unroll
        for (int i = 0; i < 8; ++i) {
            int m = i + 8 * half;
            sU[m * SU_LD + n0 + laneM]        = silu_f(acc0[i] + bias);
            sU[(m + 16) * SU_LD + n0 + laneM] = silu_f(acc1[i] + bias);
        }
    }
    __syncthreads();

    {   // stage 2: K = 128, x += u @ W2 + b2  (x-old read from sA)
        v8f acc0 = {}, acc1 = {};
        wmma_pair_k(W2, HID, sU, SU_LD, HID, n0, laneM, half, acc0, acc1);
        const float bias = b2[n0 + laneM];
#pragma unroll
        for (int i = 0; i < 8; ++i) {
            int m = i + 8 * half;
            int col = n0 + laneM;
            x[(nb0 + m) * HID + col]      = sA[m * SA_LD + col] + acc0[i] + bias;
            x[(nb0 + m + 16) * HID + col] = sA[(m + 16) * SA_LD + col] + acc1[i] + bias;
        }
    }
}

// --------------------------------------------------------- v update ----
// rows of [3N,128]: v_new = v + agg_v + v @ mixW   (32 rows per block)
__global__ void __launch_bounds__(256)
vmix_kernel(float* __restrict__ v, const float* __restrict__ agg_v,
            const float* __restrict__ W)
{
    __shared__ float sA[32 * SU_LD];

    const int tid = threadIdx.x;
    const int r0  = blockIdx.x * 32;

    for (int idx = tid; idx < 32 * HID; idx += 256) {
        int m = idx >> 7, k = idx & 127;
        sA[m * SU_LD + k] = v[(r0 + m) * HID + k];
    }
    __syncthreads();

    const int lane  = tid & 31;
    const int wave  = tid >> 5;
    const int laneM = lane & 15;
    const int half  = lane >> 4;
    const int n0    = wave * 16;

    v8f acc0 = {}, acc1 = {};
    wmma_pair_k(W, HID, sA, SU_LD, HID, n0, laneM, half, acc0, acc1);
#pragma unroll
    for (int i = 0; i < 8; ++i) {
        int m = i + 8 * half;
        int col = n0 + laneM;
        v[(r0 + m) * HID + col] =
            sA[m * SU_LD + col] + agg_v[(r0 + m) * HID + col] + acc0[i];
        v[(r0 + m + 16) * HID + col] =
            sA[(m + 16) * SU_LD + col] + agg_v[(r0 + m + 16) * HID + col] + acc1[i];
    }
}

// ---------------------------------------------------------- readout ----
__global__ void pool_kernel(const float* __restrict__ v,
                            const int* __restrict__ batch,
                            float* __restrict__ molv)
{
    int idx = blockIdx.x * blockDim.x + threadIdx.x;
    if (idx < N_NODES * 3 * HID) {
        int node = idx / (3 * HID);
        int j    = idx - node * (3 * HID);
        atomicAdd(&molv[batch[node] * (3 * HID) + j], v[idx]);
    }
}

__global__ void dip_kernel(const float* __restrict__ molv,
                           const float* __restrict__ finW,
                           float* __restrict__ out)
{
    int o = blockIdx.x * blockDim.x + threadIdx.x;   // o = mol*3 + axis
    if (o < N_MOLS * 3) {
        const float* rowp = molv + (size_t)o * HID;
        float s = 0.0f;
#pragma unroll 8
        for (int c = 0; c < HID; ++c) s += rowp[c] * finW[c];
        out[o] = s;
    }
}

// ------------------------------------------------------------ driver ----
extern "C" void kernel_launch(void* const* d_in, const int* in_sizes, int n_in,
                              void* d_out, int out_size, void* d_ws, size_t ws_size,
                              hipStream_t stream)
{
    const int*   z     = (const int*)  d_in[0];
    const float* pos   = (const float*)d_in[1];
    const int*   ei    = (const int*)  d_in[2];
    const int*   batch = (const int*)  d_in[3];
    const float* emb   = (const float*)d_in[4];
    const float* msgW1 = (const float*)d_in[5];
    const float* msgb1 = (const float*)d_in[6];
    const float* msgW2 = (const float*)d_in[7];
    const float* msgb2 = (const float*)d_in[8];
    const float* updW1 = (const float*)d_in[9];
    const float* updb1 = (const float*)d_in[10];
    const float* updW2 = (const float*)d_in[11];
    const float* updb2 = (const float*)d_in[12];
    const float* mixW  = (const float*)d_in[13];
    const float* finW  = (const float*)d_in[14];
    float* out = (float*)d_out;

    float* ws = (float*)d_ws;
    size_t o = 0;
    float* x     = ws + o; o += (size_t)N_NODES * HID;
    float* v     = ws + o; o += (size_t)N_NODES * 3 * HID;
    float* agg_s = ws + o; o += (size_t)N_NODES * HID;      // agg_s,agg_v contiguous
    float* agg_v = ws + o; o += (size_t)N_NODES * 3 * HID;
    float* dist  = ws + o; o += (size_t)N_EDGES;
    float* dir   = ws + o; o += (size_t)N_EDGES * 3;
    float* molv  = ws + o; o += (size_t)N_MOLS * 3 * HID;

    init_x_kernel<<<(N_NODES * HID + 255) / 256, 256, 0, stream>>>(z, emb, x);
    init_v_kernel<<<(N_NODES * 3 * HID + 255) / 256, 256, 0, stream>>>(pos, v);
    geom_kernel  <<<(N_EDGES + 255) / 256, 256, 0, stream>>>(pos, ei, dist, dir);

    for (int l = 0; l < 3; ++l) {
        const int aggN = N_NODES * HID + N_NODES * 3 * HID;
        zero_kernel<<<(aggN + 255) / 256, 256, 0, stream>>>(agg_s, aggN);
        edge_kernel<<<N_EDGES / 32, 256, 0, stream>>>(
            x, ei, dist, dir,
            msgW1 + (size_t)l * 257 * H2, msgb1 + (size_t)l * H2,
            msgW2 + (size_t)l * H2 * H2,  msgb2 + (size_t)l * H2,
            agg_s, agg_v);
        node_kernel<<<N_NODES / 32, 256, 0, stream>>>(
            x, agg_s,
            updW1 + (size_t)l * H2 * HID,  updb1 + (size_t)l * HID,
            updW2 + (size_t)l * HID * HID, updb2 + (size_t)l * HID);
        vmix_kernel<<<(3 * N_NODES) / 32, 256, 0, stream>>>(
            v, agg_v, mixW + (size_t)l * HID * HID);
    }

    const int mvN = N_MOLS * 3 * HID;
    zero_kernel<<<(mvN + 255) / 256, 256, 0, stream>>>(molv, mvN);
    pool_kernel<<<(N_NODES * 3 * HID + 255) / 256, 256, 0, stream>>>(v, batch, molv);
    dip_kernel <<<(N_MOLS * 3 + 255) / 256, 256, 0, stream>>>(molv, finW, out);
}